// RBFKernelProvider_73005854097654
// MI455X (gfx1250) — compile-verified
//
#include <hip/hip_runtime.h>
#include <hip/hip_bf16.h>
#include <math.h>

// Problem dims (fixed by the reference)
#define NN 4096
#define MM 4096
#define DD 8

#define F32_TINY 1.17549435082228750797e-38f   // jnp.finfo(float32).tiny
#define HALF_LOG2E 0.72134752044448170368f     // 0.5*log2(e)

typedef __attribute__((ext_vector_type(2))) float v2f;
typedef __attribute__((ext_vector_type(8))) float v8f;

// softplus(x) = log(1+exp(x)), numerically stable
__device__ __forceinline__ float softplus_f(float x) {
    return fmaxf(x, 0.0f) + log1pf(__expf(-fabsf(x)));
}

// -----------------------------------------------------------------------------
// Precompute: u = x / ls, v = x2 / ls, sqn = |u|^2, sqm = |v|^2, amp2.
// One thread per row (N + M rows). softplus of the 8 lengthscales is
// recomputed per thread (trivial vs. memory). Thread 0 also writes amp^2
// so the hot kernel never touches libm.
// -----------------------------------------------------------------------------
__global__ __launch_bounds__(256) void rbf_precompute(
    const float* __restrict__ x, const float* __restrict__ x2,
    const float* __restrict__ raw_amp, const float* __restrict__ raw_ls,
    float* __restrict__ u, float* __restrict__ v,
    float* __restrict__ sqn, float* __restrict__ sqm,
    float* __restrict__ amp2_out)
{
    int r = blockIdx.x * blockDim.x + threadIdx.x;
    if (r >= NN + MM) return;

    if (r == 0) {
        float amp = softplus_f(raw_amp[0]) + F32_TINY;
        amp2_out[0] = amp * amp;
    }

    float inv_ls[DD];
#pragma unroll
    for (int d = 0; d < DD; ++d) {
        float ls = softplus_f(raw_ls[d]) + F32_TINY;
        inv_ls[d] = 1.0f / ls;
    }

    const float* src;
    float* dst;
    float* sq;
    int row;
    if (r < NN) { row = r;      src = x  + (size_t)row * DD; dst = u + (size_t)row * DD; sq = sqn + row; }
    else        { row = r - NN; src = x2 + (size_t)row * DD; dst = v + (size_t)row * DD; sq = sqm + row; }

    float acc = 0.0f;
#pragma unroll
    for (int d = 0; d < DD; ++d) {
        float s = src[d] * inv_ls[d];
        dst[d] = s;
        acc = fmaf(s, s, acc);
    }
    *sq = acc;
}

// -----------------------------------------------------------------------------
// Main kernel: one wave32 per 16x16 output tile.
//   dot = U(16x8) * V^T(8x16) via two V_WMMA_F32_16X16X4_F32 (K=0..3, K=4..7)
//   out = amp^2 * exp2(-0.5*log2e*(sqn + sqm - 2*dot))
//
// A-fragment layout (32-bit A 16x4, ISA 7.12.2): lanes 0-15 -> M=lane, K=vgpr;
// lanes 16-31 -> M=lane-16, K=vgpr+2.  B (4x16) mirrors with N across lanes.
// C/D layout: VGPR g -> row M = g + 8*(lane>=16), col N = lane%16.
//
// Output is 64 MB written exactly once (bandwidth floor ~2.9us at 23.3 TB/s);
// u/v/sqn/sqm (~300 KB) stay resident in L2, so stores are the only HBM
// traffic. exp uses the raw v_exp_f32 (arg <= 0, flush-to-zero underflow is
// fine) to avoid libm's denormal-range fixup.
// -----------------------------------------------------------------------------
__global__ __launch_bounds__(256) void rbf_wmma_tile(
    const float* __restrict__ u, const float* __restrict__ v,
    const float* __restrict__ sqn, const float* __restrict__ sqm,
    const float* __restrict__ amp2_ws,
    float* __restrict__ out)
{
    const int lane = threadIdx.x & 31;
    const int wid  = blockIdx.x * (256 / 32) + (threadIdx.x >> 5);

    // 256 tiles per row of tiles (M/16); consecutive waves -> consecutive
    // column tiles of the same rows => contiguous store segments.
    const int tn = wid >> 8;
    const int tm = wid & 255;
    const int n0 = tn * 16;
    const int m0 = tm * 16;

    const float amp2 = amp2_ws[0];

    const int half = lane >> 4;   // 0: K=0,1 (then 4,5); 1: K=2,3 (then 6,7)
    const int l16  = lane & 15;

    const float* arow = u + (size_t)(n0 + l16) * DD + 2 * half;
    const float* brow = v + (size_t)(m0 + l16) * DD + 2 * half;

    v2f a0 = *(const v2f*)(arow);       // K = 2*half + {0,1}
    v2f a1 = *(const v2f*)(arow + 4);   // K = 4 + 2*half + {0,1}
    v2f b0 = *(const v2f*)(brow);
    v2f b1 = *(const v2f*)(brow + 4);

    v8f c = {};
    c = __builtin_amdgcn_wmma_f32_16x16x4_f32(false, a0, false, b0,
                                              (short)0, c, false, false);
    c = __builtin_amdgcn_wmma_f32_16x16x4_f32(false, a1, false, b1,
                                              (short)0, c, false, false);

    const int   col = m0 + l16;
    const float sm  = sqm[col];

#pragma unroll
    for (int g = 0; g < 8; ++g) {
        int   row = n0 + g + 8 * half;
        float s   = sqn[row] + sm - 2.0f * c[g];
        // raw v_exp_f32: argument is <= 0; underflow flushes to 0 == reference
        float r   = amp2 * __builtin_amdgcn_exp2f(-HALF_LOG2E * s);
        // 64 MB streamed once, never re-read: non-temporal store
        __builtin_nontemporal_store(r, out + (size_t)row * MM + col);
    }
}

// -----------------------------------------------------------------------------
// Launch
// -----------------------------------------------------------------------------
extern "C" void kernel_launch(void* const* d_in, const int* in_sizes, int n_in,
                              void* d_out, int out_size, void* d_ws, size_t ws_size,
                              hipStream_t stream) {
    const float* x      = (const float*)d_in[0];  // (N, D)
    const float* x2     = (const float*)d_in[1];  // (M, D)
    const float* raw_a  = (const float*)d_in[2];  // (1,)
    const float* raw_ls = (const float*)d_in[3];  // (D,)
    float*       out    = (float*)d_out;          // (N, M)

    float* ws   = (float*)d_ws;
    float* u    = ws;                        // N*D
    float* v    = u + (size_t)NN * DD;       // M*D
    float* sqn  = v + (size_t)MM * DD;       // N
    float* sqm  = sqn + NN;                  // M
    float* amp2 = sqm + MM;                  // 1
    // total: 2*4096*8 + 2*4096 + 1 floats ~= 288 KiB of workspace

    rbf_precompute<<<(NN + MM + 255) / 256, 256, 0, stream>>>(
        x, x2, raw_a, raw_ls, u, v, sqn, sqm, amp2);

    // (N/16)*(M/16) = 65536 tiles, 8 waves per 256-thread block -> 8192 blocks
    rbf_wmma_tile<<<(NN / 16) * (MM / 16) / 8, 256, 0, stream>>>(
        u, v, sqn, sqm, amp2, out);
}